// AdaptiveGraphConv_51015621542094
// MI455X (gfx1250) — compile-verified
//
#include <hip/hip_runtime.h>

typedef float v2f __attribute__((ext_vector_type(2)));
typedef float v4f __attribute__((ext_vector_type(4)));
typedef float v8f __attribute__((ext_vector_type(8)));
typedef unsigned int u32x4 __attribute__((ext_vector_type(4)));
typedef int i32x4 __attribute__((ext_vector_type(4)));
typedef int i32x8 __attribute__((ext_vector_type(8)));

#define NB   64
#define CIN  64
#define COUT 128
#define TTOT 300
#define NJ   25
#define TC   16
#define JC   400            // TC*NJ columns per slab
#define NCH  19             // ceil(300/16)
#define TN   7500           // TTOT*NJ
#define XP   76             // XaT pitch (floats): 304B, 8B-multiple, bank-conflict-free

#if defined(__has_builtin)
#if __has_builtin(__builtin_amdgcn_tensor_load_to_lds)
#define HAVE_TDM 1
#endif
#endif

#if __has_include(<hip/amd_detail/amd_gfx1250_TDM.h>)
#define TDM_SIX_ARGS 1
#endif

__device__ __forceinline__ v8f wmma4(v2f a, v2f b, v8f c) {
  // D = A(16x4 f32) * B(4x16 f32) + C(16x16 f32)
  return __builtin_amdgcn_wmma_f32_16x16x4_f32(false, a, false, b, (short)0, c, false, false);
}

__device__ __forceinline__ v8f v8zero() {
  v8f z = {0.f,0.f,0.f,0.f,0.f,0.f,0.f,0.f};
  return z;
}

#ifdef HAVE_TDM
// 2D Tensor-Data-Mover load: tile (tile_d1 rows x tile_d0 cols of f32) from
// row-major global (row stride stride0 floats) into contiguous LDS rows.
// Columns >= tensor_d0 read as zero (hardware OOB), giving free zero-padding.
__device__ __forceinline__ void tdm_load_2d(unsigned lds_byte, unsigned long long gaddr,
                                            unsigned tensor_d0, unsigned tensor_d1,
                                            unsigned tile_d0, unsigned tile_d1,
                                            unsigned long long stride0) {
  u32x4 g0;
  g0[0] = 1u;                                          // count=1, user mode
  g0[1] = lds_byte;                                    // lds_addr
  g0[2] = (unsigned)(gaddr & 0xFFFFFFFFull);           // global_addr[31:0]
  g0[3] = (unsigned)((gaddr >> 32) & 0x01FFFFFFull) | (2u << 30);  // addr[56:32] | type=2
  i32x8 g1;
  g1[0] = (int)(2u << 16);                             // data_size = 4 bytes
  g1[1] = (int)((tensor_d0 & 0xFFFFu) << 16);          // tensor_dim0[15:0] (bits 63:48)
  g1[2] = (int)((tensor_d0 >> 16) | ((tensor_d1 & 0xFFFFu) << 16));
  g1[3] = (int)((tensor_d1 >> 16) | (tile_d0 << 16));  // tile_dim0
  g1[4] = (int)(tile_d1);                              // tile_dim1 | tile_dim2(=0)<<16
  g1[5] = (int)(stride0 & 0xFFFFFFFFull);              // tensor_dim0_stride[31:0]
  g1[6] = (int)((stride0 >> 32) & 0xFFFFull);          // stride[47:32] | d1stride lo(=0)
  g1[7] = 0;
  i32x4 z4 = {0, 0, 0, 0};
#ifdef TDM_SIX_ARGS
  i32x8 z8 = {0, 0, 0, 0, 0, 0, 0, 0};
  __builtin_amdgcn_tensor_load_to_lds(g0, g1, z4, z4, z8, 0);
#else
  __builtin_amdgcn_tensor_load_to_lds(g0, g1, z4, z4, 0);
#endif
}

__device__ __forceinline__ unsigned lds_lo32(const void* p) {
  // Flat LDS pointers carry the byte offset in the low 32 bits (aperture in high bits).
  return (unsigned)(size_t)p;
}
#endif

// ---------------- Kernel 0: G = Theta^T Phi, r = Phi^T theta_b ----------------
__global__ __launch_bounds__(256) void k0_prep(const float* __restrict__ tw,
                                               const float* __restrict__ tb,
                                               const float* __restrict__ pw,
                                               float* __restrict__ G,
                                               float* __restrict__ r) {
  int tid = threadIdx.x;
  for (int idx = tid; idx < 4096; idx += 256) {
    int c = idx >> 6, c2 = idx & 63;
    float s = 0.f;
    for (int d = 0; d < 64; ++d) s += tw[d * 64 + c] * pw[d * 64 + c2];
    G[idx] = s;
  }
  if (tid < 64) {
    float s = 0.f;
    for (int d = 0; d < 64; ++d) s += pw[d * 64 + tid] * tb[d];
    r[tid] = s;
  }
}

// ---------------- Kernel 1: attention-score partials via WMMA ----------------
// Per (b, t-chunk): Y = G @ Xs  then  Spart = Xs^T(over c,t) @ Y  (32x32 padded)
__global__ __launch_bounds__(256) void k1_scores(const float* __restrict__ x,
                                                 const float* __restrict__ Gw,
                                                 const float* __restrict__ rw,
                                                 float* __restrict__ Spart,
                                                 float* __restrict__ wpart) {
  __shared__ __align__(16) float Xs[CIN][JC];
  __shared__ __align__(16) float Y[CIN][JC];
  __shared__ float Gs[64][64];
  __shared__ float rL[64];
  int tid = threadIdx.x;
  int b = blockIdx.x / NCH, ch = blockIdx.x % NCH;
  int t0 = ch * TC;
  int lane = tid & 31, wave = tid >> 5;
  int lim = TN - t0 * NJ;  // valid flattened columns in this chunk

#ifdef HAVE_TDM
  if (wave == 0) {
    unsigned long long ga = (unsigned long long)(size_t)(x + (size_t)b * CIN * TN + t0 * NJ);
    tdm_load_2d(lds_lo32(&Xs[0][0]), ga, (unsigned)lim, CIN, JC, CIN, TN);
  }
#else
  {
    const float* xb = x + (size_t)b * CIN * TN + t0 * NJ;
    for (int idx = tid; idx < CIN * JC; idx += 256) {
      int c = idx / JC, j = idx - c * JC;
      Xs[c][j] = (j < lim) ? xb[(size_t)c * TN + j] : 0.f;
    }
  }
#endif
  for (int idx = tid; idx < 4096; idx += 256) Gs[idx >> 6][idx & 63] = Gw[idx];
  if (tid < 64) rL[tid] = rw[tid];
#ifdef HAVE_TDM
  if (wave == 0) __builtin_amdgcn_s_wait_tensorcnt(0);
#endif
  __syncthreads();

  // w[m] partial (rank-1 bias term surviving softmax)
  if (tid < NJ) {
    float s = 0.f;
    for (int c = 0; c < 64; ++c) {
      float rv = rL[c];
      for (int t = 0; t < TC; ++t) s += rv * Xs[c][t * NJ + tid];
    }
    wpart[(size_t)(b * NCH + ch) * 32 + tid] = s;
  }

  int row = lane & 15, khl = lane >> 4;  // lane-half selects K pair {0,1} vs {2,3}

  // GEMM1: Y[c'][j] = sum_c Gs[c'][c] * Xs[c][j]
  {
    int rt = wave & 3, jh = wave >> 2;
    v2f areg[16];
    for (int cc = 0; cc < 16; ++cc) {
      int kk = 4 * cc + 2 * khl;
      areg[cc].x = Gs[rt * 16 + row][kk];
      areg[cc].y = Gs[rt * 16 + row][kk + 1];
    }
    int jt0 = jh ? 13 : 0, jt1 = jh ? 25 : 13;
    for (int jt = jt0; jt < jt1; ++jt) {
      v8f acc = v8zero();
      int j = jt * 16 + row;
      for (int cc = 0; cc < 16; ++cc) {
        int kk = 4 * cc + 2 * khl;
        v2f bv; bv.x = Xs[kk][j]; bv.y = Xs[kk + 1][j];
        acc = wmma4(areg[cc], bv, acc);
      }
      for (int vi = 0; vi < 8; ++vi) Y[rt * 16 + 8 * khl + vi][j] = acc[vi];
    }
  }
  __syncthreads();

  // GEMM2: Spart[n][m] = sum_{c,t} Xs[c][t*25+n] * Y[c][t*25+m]
  if (wave < 4) {
    int rt = wave >> 1, ct = wave & 1;
    int n = rt * 16 + row, m = ct * 16 + row;
    bool nv = (n < NJ), mv = (m < NJ);
    v8f acc = v8zero();
    for (int kc = 0; kc < 256; ++kc) {
      int p = 4 * kc + 2 * khl;      // flattened (c,t) pair index, even
      int c = p >> 4, t = p & 15;    // p+1 is same c, t+1
      v2f av, bv;
      av.x = nv ? Xs[c][t * NJ + n] : 0.f;
      av.y = nv ? Xs[c][(t + 1) * NJ + n] : 0.f;
      bv.x = mv ? Y[c][t * NJ + m] : 0.f;
      bv.y = mv ? Y[c][(t + 1) * NJ + m] : 0.f;
      acc = wmma4(av, bv, acc);
    }
    float* sp = Spart + (size_t)(b * NCH + ch) * 1024;
    for (int vi = 0; vi < 8; ++vi)
      sp[(rt * 16 + 8 * khl + vi) * 32 + ct * 16 + row] = acc[vi];
  }
}

// ---------------- Kernel 2: reduce + softmax + Adj build ----------------
__global__ __launch_bounds__(32) void k2_adj(const float* __restrict__ Spart,
                                             const float* __restrict__ wpart,
                                             const float* __restrict__ Afix,
                                             const float* __restrict__ Bp,
                                             float* __restrict__ Adj) {
  __shared__ float S[NJ][NJ];
  __shared__ float wsum[NJ];
  int tid = threadIdx.x, b = blockIdx.x;
  if (tid < NJ) {
    float s = 0.f;
    for (int c = 0; c < NCH; ++c) s += wpart[(size_t)(b * NCH + c) * 32 + tid];
    wsum[tid] = s;
  }
  __syncthreads();
  for (int idx = tid; idx < NJ * NJ; idx += 32) {
    int n = idx / NJ, m = idx - n * NJ;
    float s = 0.f;
    for (int c = 0; c < NCH; ++c) s += Spart[(size_t)(b * NCH + c) * 1024 + n * 32 + m];
    S[n][m] = s + wsum[m];
  }
  __syncthreads();
  if (tid < NJ) {
    float mx = -3.0e38f;
    for (int m = 0; m < NJ; ++m) mx = fmaxf(mx, S[tid][m]);
    float e[NJ], ssum = 0.f;
    for (int m = 0; m < NJ; ++m) { e[m] = expf(S[tid][m] - mx); ssum += e[m]; }
    float inv = 1.f / ssum;
    for (int m = 0; m < NJ; ++m) S[tid][m] = e[m] * inv;
  }
  __syncthreads();
  for (int idx = tid; idx < 3 * NJ * NJ; idx += 32) {
    int rem = idx % 625;
    Adj[(size_t)b * 1875 + idx] = Afix[idx] + Bp[idx] + S[rem / NJ][rem % NJ];
  }
}

// ---------------- Kernel 3: main fused (joint-mix + K convs + residual) ----------------
__global__ __launch_bounds__(512) void k3_main(const float* __restrict__ x,
                                               const float* __restrict__ Adj,
                                               const float* __restrict__ cw,
                                               const float* __restrict__ cb,
                                               const float* __restrict__ rwt,
                                               const float* __restrict__ rb,
                                               float* __restrict__ out) {
  __shared__ __align__(16) float Xs[CIN][JC];      // row-major x slab (TDM layout)
  __shared__ __align__(16) float XaT[JC][XP];      // mixed slab, transposed [j][c]
  __shared__ __align__(16) float AdjPL[3 * 700];   // [k][n-pair(14)][m(25)][2], zero padded
  __shared__ float biasL[COUT];
  int tid = threadIdx.x;
  int b = blockIdx.x / NCH, ch = blockIdx.x % NCH;
  int t0 = ch * TC;
  int lane = tid & 31, wave = tid >> 5;
  int lim = TN - t0 * NJ;

#ifdef HAVE_TDM
  if (wave == 0) {
    unsigned long long ga = (unsigned long long)(size_t)(x + (size_t)b * CIN * TN + t0 * NJ);
    tdm_load_2d(lds_lo32(&Xs[0][0]), ga, (unsigned)lim, CIN, JC, CIN, TN);
  }
#else
  {
    const float* xb = x + (size_t)b * CIN * TN + t0 * NJ;
    for (int idx = tid; idx < CIN * JC; idx += 512) {
      int c = idx / JC, j = idx - c * JC;
      Xs[c][j] = (j < lim) ? xb[(size_t)c * TN + j] : 0.f;
    }
  }
#endif
  // Gather Adj into pair-interleaved layout: AdjPL[k*700 + (n>>1)*50 + m*2 + (n&1)],
  // with zeros wherever n >= 25 (so K-pad needs no per-fetch guard on B).
  for (int idx = tid; idx < 3 * 700; idx += 512) {
    int k = idx / 700, rr = idx - k * 700;
    int p = rr / 50, q = rr - p * 50;
    int m = q >> 1, n = 2 * p + (q & 1);
    AdjPL[idx] = (n < NJ) ? Adj[(size_t)b * 1875 + k * 625 + n * NJ + m] : 0.f;
  }
  if (tid < COUT) biasL[tid] = cb[tid] + cb[COUT + tid] + cb[2 * COUT + tid] + rb[tid];
#ifdef HAVE_TDM
  if (wave == 0) __builtin_amdgcn_s_wait_tensorcnt(0);
#endif
  __syncthreads();

  int row = lane & 15, khl = lane >> 4;
  int dt = wave & 7, jh = wave >> 3;
  int jt0 = jh ? 13 : 0, jtn = jh ? 12 : 13;
  int d = dt * 16 + row;

  v8f acc[13];
  for (int i = 0; i < 13; ++i) acc[i] = v8zero();
  v2f wreg[16];

  // Residual pass: A = res_w rows [d-tile], B = Xs (identity adjacency)
  for (int cc = 0; cc < 16; ++cc) {
    int kk = 4 * cc + 2 * khl;
    wreg[cc].x = rwt[d * 64 + kk];
    wreg[cc].y = rwt[d * 64 + kk + 1];
  }
  for (int i = 0; i < jtn; ++i) {
    int j = (jt0 + i) * 16 + row;
    for (int cc = 0; cc < 16; ++cc) {
      int kk = 4 * cc + 2 * khl;
      v2f bv; bv.x = Xs[kk][j]; bv.y = Xs[kk + 1][j];
      acc[i] = wmma4(wreg[cc], bv, acc[i]);
    }
  }

  for (int k = 0; k < 3; ++k) {
    __syncthreads();  // prior readers of XaT done before overwrite
    // Joint mixing via WMMA: XaT[t*25+m][c] = sum_n Xs[c][t*25+n] * Adj[k][n][m]
    // rows = 1024 flattened (c,t) pairs -> 64 row tiles; cols = 25 (2 tiles); K = 25 -> 7 chunks
    const float* adjk = &AdjPL[k * 700];
    for (int it = 0; it < 8; ++it) {
      int ti = wave + 16 * it;          // 0..127
      int rtile = ti >> 1, ctile = ti & 1;
      int c = rtile;                    // channel for this 16-row tile (t = row)
      int m = ctile * 16 + row;
      bool mvv = (m < NJ);
      v8f a2 = v8zero();
      for (int nc = 0; nc < 7; ++nc) {
        int kk = 4 * nc + 2 * khl;      // n index pair (even)
        v2f av;
        av.x = (kk < NJ) ? Xs[c][row * NJ + kk] : 0.f;
        av.y = (kk + 1 < NJ) ? Xs[c][row * NJ + kk + 1] : 0.f;
        v2f bv = *(const v2f*)(adjk + (2 * nc + khl) * 50 + m * 2);  // zeros pad n>=25
        a2 = wmma4(av, bv, a2);
      }
      if (mvv) {
        for (int vi = 0; vi < 8; ++vi) {
          int t2 = 8 * khl + vi;
          XaT[t2 * NJ + m][c] = a2[vi];
        }
      }
    }
    __syncthreads();
    const float* wk = cw + (size_t)k * COUT * 64;
    for (int cc = 0; cc < 16; ++cc) {
      int kk = 4 * cc + 2 * khl;
      wreg[cc].x = wk[d * 64 + kk];
      wreg[cc].y = wk[d * 64 + kk + 1];
    }
    for (int i = 0; i < jtn; ++i) {
      int j = (jt0 + i) * 16 + row;
      const float* bp = &XaT[j][0];     // one base per j-tile; K offsets are DS immediates
      for (int cc = 0; cc < 16; ++cc) {
        int kk = 4 * cc + 2 * khl;
        v2f bv = *(const v2f*)(bp + kk);  // single aligned ds_load_b64
        acc[i] = wmma4(wreg[cc], bv, acc[i]);
      }
    }
  }

  // Store pre-BN output + bias
  int jlim = lim < JC ? lim : JC;
  float* ob = out + (size_t)b * COUT * TN + t0 * NJ;
  for (int i = 0; i < jtn; ++i) {
    int j = (jt0 + i) * 16 + row;
    if (j < jlim) {
      for (int vi = 0; vi < 8; ++vi) {
        int dd = dt * 16 + 8 * khl + vi;
        ob[(size_t)dd * TN + j] = acc[i][vi] + biasL[dd];
      }
    }
  }
}

// ---------------- Kernel 4: per-channel batch stats (deterministic tree) ----------------
__global__ __launch_bounds__(256) void k4_stats(const float* __restrict__ o,
                                                const float* __restrict__ gamma,
                                                const float* __restrict__ beta,
                                                float* __restrict__ scale,
                                                float* __restrict__ shift) {
  __shared__ float sb[256], s2b[256];
  int tid = threadIdx.x, d = blockIdx.x;
  float s = 0.f, s2 = 0.f;
  for (int b = 0; b < NB; ++b) {
    const v4f* p = (const v4f*)(o + (size_t)(b * COUT + d) * TN);
    for (int i = tid; i < TN / 4; i += 256) {
      v4f v = p[i];
      s += v.x + v.y + v.z + v.w;
      s2 += v.x * v.x + v.y * v.y + v.z * v.z + v.w * v.w;
    }
  }
  sb[tid] = s; s2b[tid] = s2;
  __syncthreads();
  for (int off = 128; off > 0; off >>= 1) {
    if (tid < off) { sb[tid] += sb[tid + off]; s2b[tid] += s2b[tid + off]; }
    __syncthreads();
  }
  if (tid == 0) {
    float cnt = (float)(NB * TN);
    float mean = sb[0] / cnt;
    float var = s2b[0] / cnt - mean * mean;
    float sc = gamma[d] * rsqrtf(fmaxf(var, 0.f) + 1e-5f);
    scale[d] = sc;
    shift[d] = beta[d] - mean * sc;
  }
}

// ---------------- Kernel 5: in-place BN + ReLU (128-bit streams) ----------------
__global__ __launch_bounds__(256) void k5_bnrelu(float* __restrict__ o,
                                                 const float* __restrict__ scale,
                                                 const float* __restrict__ shift,
                                                 int total4) {
  int i = blockIdx.x * 256 + threadIdx.x;
  if (i < total4) {
    // 7500 % 4 == 0 -> a float4 never straddles a (b,d) row; d uniform across it
    int d = ((i * 4) / TN) & (COUT - 1);
    v4f* p = (v4f*)o;
    v4f v = p[i];
    float sc = scale[d], sh = shift[d];
    v.x = fmaxf(v.x * sc + sh, 0.f);
    v.y = fmaxf(v.y * sc + sh, 0.f);
    v.z = fmaxf(v.z * sc + sh, 0.f);
    v.w = fmaxf(v.w * sc + sh, 0.f);
    p[i] = v;
  }
}

extern "C" void kernel_launch(void* const* d_in, const int* in_sizes, int n_in,
                              void* d_out, int out_size, void* d_ws, size_t ws_size,
                              hipStream_t stream) {
  (void)in_sizes; (void)n_in; (void)out_size; (void)ws_size;
  const float* x     = (const float*)d_in[0];
  const float* Afix  = (const float*)d_in[1];
  const float* tw    = (const float*)d_in[2];
  const float* tb    = (const float*)d_in[3];
  const float* pw    = (const float*)d_in[4];
  const float* pb    = (const float*)d_in[5]; (void)pb;  // cancels in softmax
  const float* Bp    = (const float*)d_in[6];
  const float* cw    = (const float*)d_in[7];
  const float* cb    = (const float*)d_in[8];
  const float* rwt   = (const float*)d_in[9];
  const float* rb    = (const float*)d_in[10];
  const float* gamma = (const float*)d_in[11];
  const float* beta  = (const float*)d_in[12];
  float* out = (float*)d_out;
  float* ws  = (float*)d_ws;

  float* G     = ws;                                   // 4096
  float* rvec  = ws + 4096;                            // 64 (pad 128)
  float* Spart = ws + 4096 + 128;                      // 64*19*1024
  float* wpart = Spart + (size_t)NB * NCH * 1024;      // 64*19*32
  float* AdjW  = wpart + (size_t)NB * NCH * 32;        // 64*1875
  float* scale = AdjW + (size_t)NB * 1875;             // 128
  float* shift = scale + 128;                          // 128

  hipLaunchKernelGGL(k0_prep, dim3(1), dim3(256), 0, stream, tw, tb, pw, G, rvec);
  hipLaunchKernelGGL(k1_scores, dim3(NB * NCH), dim3(256), 0, stream, x, G, rvec, Spart, wpart);
  hipLaunchKernelGGL(k2_adj, dim3(NB), dim3(32), 0, stream, Spart, wpart, Afix, Bp, AdjW);
  hipLaunchKernelGGL(k3_main, dim3(NB * NCH), dim3(512), 0, stream, x, AdjW, cw, cb, rwt, rb, out);
  hipLaunchKernelGGL(k4_stats, dim3(COUT), dim3(256), 0, stream, out, gamma, beta, scale, shift);
  int total4 = NB * COUT * TN / 4;
  hipLaunchKernelGGL(k5_bnrelu, dim3((total4 + 255) / 256), dim3(256), 0, stream, out, scale, shift, total4);
}